// GridSmoother_65944927863084
// MI455X (gfx1250) — compile-verified
//
#include <hip/hip_runtime.h>

// ---------------------------------------------------------------------------
// Block-CG solver for L x = b, L = I + Dx^T Wx Dx + Dy^T Wy Dy on a 48x64
// grid, 16 RHS (channels), 4 batches.  One workgroup (32 wave32's) per batch,
// everything LDS/register resident.  16x16 block algebra done with
// v_wmma_f32_16x16x4_f32.
// ---------------------------------------------------------------------------

typedef float v2f __attribute__((ext_vector_type(2)));
typedef float v8f __attribute__((ext_vector_type(8)));

#define HWPX  3072        // 48*64 pixels
#define HPX   48
#define WPX   64
#define NCH   16
#define PS    18          // padded row stride (floats) for P in LDS
#define NITER 32          // kappa(L) <= ~9 -> 0.5^32 ~ 2e-10 residual

// D = A(16x4) * B(4x16) + C(16x16), all f32.
static __device__ __forceinline__ v8f wmma_k4(v2f a, v2f b, v8f c) {
  return __builtin_amdgcn_wmma_f32_16x16x4_f32(false, a, false, b, (short)0, c,
                                               false, false);
}

// SWAPX16 lane exchange (group-of-32 swizzle: xor=0x10, and=0x1f)
static __device__ __forceinline__ float swapx16(float x) {
  return __int_as_float(__builtin_amdgcn_ds_swizzle(__float_as_int(x), 0x401f));
}

// Build the 4x16 B-fragment (rows 4j..4j+3) of a 16x16 tile held in WMMA
// C/D register layout (lane%16 = column, VGPR r = row r + 8*(lane/16)).
// Lane needs rows {4j+2*h2, 4j+2*h2+1}; half of those live in the partner
// half-wave -> ds_swizzle SWAPX16 + select.
static __device__ __forceinline__ v2f rtile_frag(v8f R, int j, int h2) {
  const int  base  = (j & 1) * 4;
  const bool ownLo = (j < 2);                 // which half owns the rows
  const int  ro    = base + (ownLo ? 0 : 2);  // regs if we own them
  const int  rs    = base + (ownLo ? 2 : 0);  // regs to fetch from partner
  float s0 = swapx16(R[rs]);                  // executed by all lanes
  float s1 = swapx16(R[rs + 1]);
  bool useOwn = ((h2 == 0) == ownLo);
  v2f f;
  f.x = useOwn ? R[ro]     : s0;
  f.y = useOwn ? R[ro + 1] : s1;
  return f;
}

// Lockstep Gauss-Jordan on a 16x16 system with 16 RHS columns; executed by
// lanes tid<16 of wave 0 (same wave => DS ops in order, lanes in lockstep).
static __device__ void gauss16(float* M, float* Z, int j) {
  for (int k = 0; k < 16; ++k) {
    float inv = 1.0f / M[k * 16 + k];
    float fc[16];
#pragma unroll
    for (int i = 0; i < 16; ++i) fc[i] = M[i * 16 + k];  // snapshot col k
    float pk = M[k * 16 + j] * inv;
    float zk = Z[k * 16 + j] * inv;
    M[k * 16 + j] = pk;
    Z[k * 16 + j] = zk;
#pragma unroll
    for (int i = 0; i < 16; ++i) {
      if (i == k) continue;
      M[i * 16 + j] -= fc[i] * pk;
      Z[i * 16 + j] -= fc[i] * zk;
    }
  }
}

__global__ __launch_bounds__(1024, 1)
void gridsmoother_blockcg(const float* __restrict__ ae,
                          const float* __restrict__ wxy,
                          float* __restrict__ out) {
  extern __shared__ float smem[];
  float* Pb     = smem;                 // HWPX*PS   : search direction P
  float* wEs    = Pb + HWPX * PS;       // HWPX      : effective edge weights
  float* wWs    = wEs + HWPX;
  float* wSs    = wWs + HWPX;
  float* wNs    = wSs + HWPX;
  float* dgs    = wNs + HWPX;           // diagonal of L
  float* Gtmp   = dgs + HWPX;           // 256 : scratch Gram
  float* Grr    = Gtmp + 256;           // 256 : persistent R^T R
  float* alphaM = Grr + 256;            // 256
  float* betaM  = alphaM + 256;         // 256
  float* Msol   = betaM + 256;          // 256 : solver scratch

  const int tid = threadIdx.x;
  const int b   = blockIdx.x;
  const int ln  = tid & 31;
  const int n   = ln & 15;        // channel lane
  const int h2  = ln >> 4;        // half-wave
  const int pbase = 96 * (tid >> 5);   // 96 pixels (6 tiles of 16) per wave

  const float* aeb = ae  + (size_t)b * NCH * HWPX;
  const float* wxb = wxy + (size_t)b * 2 * HWPX;

  // ---- stage b = ae into P (P0 = R0 = b), coalesced global reads ----------
  for (int k = 0; k < 48; ++k) {
    int idx = tid + k * 1024;           // enumerates (c,i)
    int c = idx / HWPX;
    int i = idx - c * HWPX;
    Pb[i * PS + c] = aeb[idx];
  }
  // ---- precompute effective stencil weights + diagonal --------------------
  for (int k = 0; k < 3; ++k) {
    int i = tid + k * 1024;
    int h = i >> 6, w = i & 63;
    float wxv = wxb[i];
    float wyv = wxb[HWPX + i];
    float we = (w < WPX - 1) ? wxv : 0.f;
    float ww = (w > 0) ? wxb[i - 1] : 0.f;
    float ws = (h < HPX - 1) ? wyv : 0.f;
    float wn = (h > 0) ? wxb[HWPX + i - WPX] : 0.f;
    wEs[i] = we; wWs[i] = ww; wSs[i] = ws; wNs[i] = wn;
    dgs[i] = 1.f + we + ww + ws + wn;
  }
  __syncthreads();

  // stencil: (L p)(i, c), neighbor indices clamped (their weights are 0)
  auto qv = [&](int i, int c) -> float {
    int iE = min(i + 1, HWPX - 1);
    int iW = max(i - 1, 0);
    int iS = min(i + WPX, HWPX - 1);
    int iN = max(i - WPX, 0);
    return dgs[i] * Pb[i * PS + c] - wEs[i] * Pb[iE * PS + c] -
           wWs[i] * Pb[iW * PS + c] - wSs[i] * Pb[iS * PS + c] -
           wNs[i] * Pb[iN * PS + c];
  };

  // ---- X = 0, R = b (WMMA C/D register layout) ----------------------------
  v8f Xr[6], Rr[6];
#pragma unroll
  for (int t = 0; t < 6; ++t) {
    v8f r = {};
    v8f z = {};
#pragma unroll
    for (int rr = 0; rr < 8; ++rr) {
      int i = pbase + 16 * t + rr + 8 * h2;
      r[rr] = Pb[i * PS + n];
    }
    Rr[t] = r;
    Xr[t] = z;
  }

  // ---- Grr = R^T R --------------------------------------------------------
  if (tid < 256) Grr[tid] = 0.f;
  __syncthreads();
  {
    v8f acc = {};
#pragma unroll
    for (int t = 0; t < 6; ++t)
#pragma unroll
      for (int j = 0; j < 4; ++j) {
        v2f f = rtile_frag(Rr[t], j, h2);
        acc = wmma_k4(f, f, acc);      // A == B == R tile fragment
      }
#pragma unroll
    for (int r = 0; r < 8; ++r) atomicAdd(&Grr[(r + 8 * h2) * 16 + n], acc[r]);
  }
  __syncthreads();

  // =========================== block-CG loop ===============================
  for (int iter = 0; iter < NITER; ++iter) {
    // ---- Mpq = P^T (L P) --------------------------------------------------
    if (tid < 256) Gtmp[tid] = 0.f;
    __syncthreads();
    {
      v8f acc = {};
#pragma unroll
      for (int t = 0; t < 6; ++t) {
        int i0 = pbase + 16 * t;
#pragma unroll
        for (int j = 0; j < 4; ++j) {
          int klo = i0 + 4 * j + 2 * h2;
          v2f aT;                              // A = P^T (transpose read)
          aT.x = Pb[klo * PS + n];
          aT.y = Pb[(klo + 1) * PS + n];
          v2f bq;                              // B = Q rows (stencil on the fly)
          bq.x = qv(klo, n);
          bq.y = qv(klo + 1, n);
          acc = wmma_k4(aT, bq, acc);
        }
      }
#pragma unroll
      for (int r = 0; r < 8; ++r)
        atomicAdd(&Gtmp[(r + 8 * h2) * 16 + n], acc[r]);
    }
    __syncthreads();

    // ---- alpha = Mpq^{-1} Grr --------------------------------------------
    if (tid < 256) { Msol[tid] = Gtmp[tid]; alphaM[tid] = Grr[tid]; }
    __syncthreads();
    if (tid < 16) gauss16(Msol, alphaM, tid);
    __syncthreads();

    // ---- X += P*alpha ; R -= (L P)*alpha ---------------------------------
    {
      v2f bal[4], baln[4];
#pragma unroll
      for (int j = 0; j < 4; ++j) {            // alpha fragments (tile-invariant)
        int klo = 4 * j + 2 * h2;
        bal[j].x = alphaM[klo * 16 + n];
        bal[j].y = alphaM[(klo + 1) * 16 + n];
        baln[j].x = -bal[j].x;                 // f32 WMMA has no A/B negate
        baln[j].y = -bal[j].y;
      }
#pragma unroll
      for (int t = 0; t < 6; ++t) {
        int i0 = pbase + 16 * t;
        int ip = i0 + n;                       // this lane's A-fragment row
        int iE = min(ip + 1, HWPX - 1), iW = max(ip - 1, 0);
        int iS = min(ip + WPX, HWPX - 1), iN = max(ip - WPX, 0);
        float dv = dgs[ip], we = wEs[ip], ww = wWs[ip], ws = wSs[ip],
              wn = wNs[ip];
        v8f x = Xr[t], rr = Rr[t];
#pragma unroll
        for (int j = 0; j < 4; ++j) {
          int klo = 4 * j + 2 * h2;
          v2f ap = *(const v2f*)(Pb + ip * PS + klo);  // 8B-aligned b64 load
          x = wmma_k4(ap, bal[j], x);
          v2f aq;                              // Q row fragment via stencil
          aq.x = dv * Pb[ip * PS + klo] - we * Pb[iE * PS + klo] -
                 ww * Pb[iW * PS + klo] - ws * Pb[iS * PS + klo] -
                 wn * Pb[iN * PS + klo];
          aq.y = dv * Pb[ip * PS + klo + 1] - we * Pb[iE * PS + klo + 1] -
                 ww * Pb[iW * PS + klo + 1] - ws * Pb[iS * PS + klo + 1] -
                 wn * Pb[iN * PS + klo + 1];
          rr = wmma_k4(aq, baln[j], rr);
        }
        Xr[t] = x;
        Rr[t] = rr;
      }
    }
    __syncthreads();

    // ---- Mrr_new = R^T R --------------------------------------------------
    if (tid < 256) Gtmp[tid] = 0.f;
    __syncthreads();
    {
      v8f acc = {};
#pragma unroll
      for (int t = 0; t < 6; ++t)
#pragma unroll
        for (int j = 0; j < 4; ++j) {
          v2f f = rtile_frag(Rr[t], j, h2);
          acc = wmma_k4(f, f, acc);
        }
#pragma unroll
      for (int r = 0; r < 8; ++r)
        atomicAdd(&Gtmp[(r + 8 * h2) * 16 + n], acc[r]);
    }
    __syncthreads();

    // ---- beta = Grr^{-1} Mrr_new ; Grr <- Mrr_new -------------------------
    if (tid < 256) { Msol[tid] = Grr[tid]; betaM[tid] = Gtmp[tid]; }
    __syncthreads();
    if (tid < 16) gauss16(Msol, betaM, tid);
    __syncthreads();
    if (tid < 256) Grr[tid] = Gtmp[tid];

    // ---- P = R + P*beta (each wave owns its rows; no cross-wave hazard) ---
    {
      v2f bb[4];
#pragma unroll
      for (int j = 0; j < 4; ++j) {
        int klo = 4 * j + 2 * h2;
        bb[j].x = betaM[klo * 16 + n];
        bb[j].y = betaM[(klo + 1) * 16 + n];
      }
#pragma unroll
      for (int t = 0; t < 6; ++t) {
        int i0 = pbase + 16 * t;
        int ip = i0 + n;
        v8f acc = Rr[t];
#pragma unroll
        for (int j = 0; j < 4; ++j) {
          int klo = 4 * j + 2 * h2;
          v2f ap = *(const v2f*)(Pb + ip * PS + klo);
          acc = wmma_k4(ap, bb[j], acc);
        }
#pragma unroll
        for (int r = 0; r < 8; ++r)
          Pb[(i0 + r + 8 * h2) * PS + n] = acc[r];
      }
    }
    __syncthreads();   // P writes visible before next iteration's stencil
  }

  // ---- write X out: stage through LDS, then coalesced global stores -------
#pragma unroll
  for (int t = 0; t < 6; ++t)
#pragma unroll
    for (int r = 0; r < 8; ++r)
      Pb[(pbase + 16 * t + r + 8 * h2) * PS + n] = Xr[t][r];
  __syncthreads();
  float* outb = out + (size_t)b * NCH * HWPX;
  for (int k = 0; k < 48; ++k) {
    int idx = tid + k * 1024;
    int c = idx / HWPX;
    int i = idx - c * HWPX;
    outb[idx] = Pb[i * PS + c];
  }
}

extern "C" void kernel_launch(void* const* d_in, const int* in_sizes, int n_in,
                              void* d_out, int out_size, void* d_ws,
                              size_t ws_size, hipStream_t stream) {
  (void)in_sizes; (void)n_in; (void)out_size; (void)d_ws; (void)ws_size;
  const float* ae  = (const float*)d_in[0];   // (4,16,48,64) f32
  const float* wxy = (const float*)d_in[1];   // (4,2,48,64)  f32
  float* out = (float*)d_out;                 // (4,16,48,64) f32
  const size_t shmem = (size_t)(HWPX * PS + 5 * HWPX + 5 * 256) * sizeof(float);
  gridsmoother_blockcg<<<dim3(4), dim3(1024), shmem, stream>>>(ae, wxy, out);
}